// CausalMultiHeadAttention_42494406426963
// MI455X (gfx1250) — compile-verified
//
#include <hip/hip_runtime.h>

#define BB 4
#define SS 4096
#define DD 256
#define HH 4
#define KK 64
#define NROWS (BB * SS)                 // 16384
#define NMAT  ((size_t)NROWS * DD)      // 4194304 floats per matrix

typedef float v2f __attribute__((ext_vector_type(2)));
typedef float v4f __attribute__((ext_vector_type(4)));
typedef float v8f __attribute__((ext_vector_type(8)));
typedef int   v4i __attribute__((ext_vector_type(4)));

typedef __attribute__((address_space(1))) v4i* as1_v4i_p;   // global
typedef __attribute__((address_space(3))) v4i* as3_v4i_p;   // LDS

#if __has_builtin(__builtin_amdgcn_global_load_async_to_lds_b128)
#define USE_ASYNC_LDS 1
#else
#define USE_ASYNC_LDS 0
#endif

// D = A(16x4) * B(4x16) + C(16x16), all fp32 — native CDNA5 fp32 matrix op.
__device__ __forceinline__ v8f wmma4(v2f a, v2f b, v8f c) {
  return __builtin_amdgcn_wmma_f32_16x16x4_f32(false, a, false, b, (short)0, c,
                                               false, false);
}

// 16B memory -> LDS copy, async (ASYNCcnt path) when available.
__device__ __forceinline__ void cp16_to_lds(const float* gsrc, float* ldst) {
#if USE_ASYNC_LDS
  __builtin_amdgcn_global_load_async_to_lds_b128((as1_v4i_p)gsrc, (as3_v4i_p)ldst,
                                                 0, 0);
#else
  *(v4f*)ldst = *(const v4f*)gsrc;
#endif
}

__device__ __forceinline__ void wait_async_le4() {
#if USE_ASYNC_LDS
  asm volatile("s_wait_asynccnt 0x4" ::: "memory");
#endif
}
__device__ __forceinline__ void wait_async_le0() {
#if USE_ASYNC_LDS
  asm volatile("s_wait_asynccnt 0x0" ::: "memory");
#endif
}

// ---------------------------------------------------------------------------
// Kernel 1: fused Q/K/V projection.  [16384,256] x [256,256] GEMM per matrix.
// grid = (1024 row-tiles, 3 matrices), block = 128 (4 waves, 64 cols/wave).
// ---------------------------------------------------------------------------
__global__ void qkv_proj_kernel(const float* __restrict__ query,
                                const float* __restrict__ value,
                                const float* __restrict__ Wq, const float* __restrict__ bq,
                                const float* __restrict__ Wk, const float* __restrict__ bk,
                                const float* __restrict__ Wv, const float* __restrict__ bv,
                                float* __restrict__ qb, float* __restrict__ kb,
                                float* __restrict__ vb) {
  const int tid  = threadIdx.x;
  const int w    = tid >> 5;
  const int lane = tid & 31;
  const int l16  = lane & 15;   // row (A) / col (B,C) within tile
  const int g    = lane >> 4;   // K-pair select (A,B) / row-half (C)
  const int row0 = blockIdx.x * 16;
  const int which = blockIdx.y;

  const float* src  = (which == 0) ? query : value;
  const float* W    = (which == 0) ? Wq : (which == 1) ? Wk : Wv;
  const float* bias = (which == 0) ? bq : (which == 1) ? bk : bv;
  float*       dst  = (which == 0) ? qb : (which == 1) ? kb : vb;

  const int col0 = w * 64;
  v8f acc[4] = {v8f{}, v8f{}, v8f{}, v8f{}};
  const float* arow = src + (size_t)(row0 + l16) * DD;

  #pragma unroll 4
  for (int d = 0; d < DD; d += 4) {
    v2f af = *(const v2f*)&arow[d + 2 * g];                 // A frag (m, d+2g..+1)
    const float* wr0 = W + (size_t)(d + 2 * g) * DD + col0 + l16;
    const float* wr1 = wr0 + DD;
    #pragma unroll
    for (int t = 0; t < 4; ++t) {
      v2f bf;
      bf[0] = wr0[t * 16];                                  // B(2g+0, n)
      bf[1] = wr1[t * 16];                                  // B(2g+1, n)
      acc[t] = wmma4(af, bf, acc[t]);
    }
  }

  #pragma unroll
  for (int t = 0; t < 4; ++t) {
    const int col = col0 + t * 16 + l16;
    const float bia = bias[col];
    #pragma unroll
    for (int j = 0; j < 8; ++j) {
      const int row = row0 + j + 8 * g;                     // C layout: M = j + 8*half
      dst[(size_t)row * DD + col] = acc[t][j] + bia;
    }
  }
}

// ---------------------------------------------------------------------------
// Kernel 2: causal flash-attention per (b,h), double-buffered async staging.
// grid = (S/64, H, B), block = 128 (4 waves, 16 query rows each).
// Waves 0-1 stage the K tile, waves 2-3 stage the V tile (4 x b128/lane each).
// ---------------------------------------------------------------------------
__global__ void attn_kernel(const float* __restrict__ qb,
                            const float* __restrict__ kb,
                            const float* __restrict__ vb,
                            float* __restrict__ ctx) {
  __shared__ float Kt[2][16][68];     // K tiles, row-major, padded
  __shared__ float Vt[2][16][68];     // V tiles, row-major, padded
  __shared__ float Pst[4][16][18];    // per-wave P staging (C->A layout shuffle)

  const int tid  = threadIdx.x;
  const int w    = tid >> 5;
  const int lane = tid & 31;
  const int l16  = lane & 15;
  const int g    = lane >> 4;
  const int h    = blockIdx.y;
  const int b    = blockIdx.z;
  const int qblk = blockIdx.x;
  const int q0   = qblk * 64 + w * 16;

  // Preload this wave's Q fragments (16x64, as 16 A-frags of K=4).
  const float* qrow = qb + ((size_t)(b * SS + q0 + l16) * HH + h) * KK;
  v2f qf[16];
  #pragma unroll
  for (int t = 0; t < 16; ++t) qf[t] = *(const v2f*)&qrow[4 * t + 2 * g];

  v8f o[4] = {v8f{}, v8f{}, v8f{}, v8f{}};
  float rmax[8], rsum[8];
  #pragma unroll
  for (int j = 0; j < 8; ++j) { rmax[j] = -__builtin_inff(); rsum[j] = 0.0f; }

  const int jt_end = qblk * 4 + 3;    // block's last key tile (inclusive)
  const int my_jt  = q0 >> 4;         // this wave's last needed key tile

  // staging assignment: wave pair selects K vs V; 4 b128 chunks per wave/tile
  const float* stage_src = (w < 2) ? kb : vb;
  float (*stage_dst)[16][68] = (w < 2) ? Kt : Vt;
  const int wh = w & 1;               // which half of the tile this wave stages
  const int sr = l16;                 // staged row 0..15
  const int scb = g * 4;              // +0 / +4 within each 8-float chunk

  // prologue: stage tile 0 into buffer 0
  {
    const float* srow = stage_src + ((size_t)(b * SS + 0 + sr) * HH + h) * KK;
    #pragma unroll
    for (int q2 = 0; q2 < 4; ++q2) {
      const int c = (wh * 4 + q2) * 8 + scb;                // multiple of 4
      cp16_to_lds(&srow[c], &stage_dst[0][sr][c]);
    }
  }

  int buf = 0;
  for (int jt = 0; jt <= jt_end; ++jt) {
    const int kbase = jt * 16;
    if (jt < jt_end) {
      // prefetch next tile into the other buffer (safe: freed by prior barrier)
      const float* srow =
          stage_src + ((size_t)(b * SS + kbase + 16 + sr) * HH + h) * KK;
      #pragma unroll
      for (int q2 = 0; q2 < 4; ++q2) {
        const int c = (wh * 4 + q2) * 8 + scb;
        cp16_to_lds(&srow[c], &stage_dst[buf ^ 1][sr][c]);
      }
      wait_async_le4();               // current tile's 4 ops done (in-order)
    } else {
      wait_async_le0();
    }
    __syncthreads();

    if (jt <= my_jt) {                // wave-uniform: EXEC stays all-ones
      // S = Q * K^T  (B(k,n) = Ktile[n][k] -> contiguous b64 from Kt row).
      v8f sacc = {};
      #pragma unroll
      for (int t = 0; t < 16; ++t) {
        v2f bf = *(const v2f*)&Kt[buf][l16][4 * t + 2 * g];
        sacc = wmma4(qf[t], bf, sacc);
      }
      // Scale + additive causal mask + online softmax (rows live in
      // (VGPR j, lane-half g); reduce across the 16 lanes of the half).
      #pragma unroll
      for (int j = 0; j < 8; ++j) {
        const int qidx = q0 + j + 8 * g;
        const int kidx = kbase + l16;
        float s = sacc[j] * 0.125f + ((qidx >= kidx) ? 0.0f : -1e9f);
        float tm = s;
        #pragma unroll
        for (int off = 1; off < 16; off <<= 1) tm = fmaxf(tm, __shfl_xor(tm, off, 16));
        const float nm   = fmaxf(rmax[j], tm);
        const float corr = __expf(rmax[j] - nm);
        const float p    = __expf(s - nm);
        float ps = p;
        #pragma unroll
        for (int off = 1; off < 16; off <<= 1) ps += __shfl_xor(ps, off, 16);
        rsum[j] = rsum[j] * corr + ps;
        rmax[j] = nm;
        o[0][j] *= corr; o[1][j] *= corr; o[2][j] *= corr; o[3][j] *= corr;
        Pst[w][j + 8 * g][l16] = p;   // C-layout -> LDS for A-layout reload
      }
      // DS is in-order per wave; this stops compiler reordering of the
      // cross-lane store->load through LDS.
      asm volatile("s_wait_dscnt 0x0" ::: "memory");
      // O += P * V   (A from Pst b64; B = two b32 rows of the V tile).
      #pragma unroll
      for (int t2 = 0; t2 < 4; ++t2) {
        v2f af = *(const v2f*)&Pst[w][l16][4 * t2 + 2 * g];
        #pragma unroll
        for (int u = 0; u < 4; ++u) {
          v2f bf;
          bf[0] = Vt[buf][4 * t2 + 2 * g][u * 16 + l16];
          bf[1] = Vt[buf][4 * t2 + 2 * g + 1][u * 16 + l16];
          o[u] = wmma4(af, bf, o[u]);
        }
      }
    }
    __syncthreads();                  // all reads of buf done before reuse
    buf ^= 1;
  }

  // Normalize and store context [B,S,H,K].
  #pragma unroll
  for (int j = 0; j < 8; ++j) {
    const float inv = 1.0f / rsum[j];
    const int srow = q0 + j + 8 * g;
    float* crow = ctx + ((size_t)(b * SS + srow) * HH + h) * KK;
    #pragma unroll
    for (int u = 0; u < 4; ++u) crow[u * 16 + l16] = o[u][j] * inv;
  }
}

// ---------------------------------------------------------------------------
// Kernel 3: output projection + bias + residual + LayerNorm.
// grid = 1024 row-tiles, block = 256 (8 waves x 32 cols = full 256-wide row).
// ---------------------------------------------------------------------------
__global__ void out_ln_kernel(const float* __restrict__ ctx,
                              const float* __restrict__ Wo,
                              const float* __restrict__ bo,
                              const float* __restrict__ query,
                              const float* __restrict__ gamma,
                              const float* __restrict__ beta,
                              float* __restrict__ out) {
  __shared__ float X[16][260];

  const int tid  = threadIdx.x;
  const int w    = tid >> 5;
  const int lane = tid & 31;
  const int l16  = lane & 15;
  const int g    = lane >> 4;
  const int row0 = blockIdx.x * 16;
  const int col0 = w * 32;

  v8f acc0 = {}, acc1 = {};
  const float* arow = ctx + (size_t)(row0 + l16) * DD;

  #pragma unroll 4
  for (int d = 0; d < DD; d += 4) {
    v2f af = *(const v2f*)&arow[d + 2 * g];
    const float* wr0 = Wo + (size_t)(d + 2 * g) * DD + col0 + l16;
    const float* wr1 = wr0 + DD;
    v2f bf0; bf0[0] = wr0[0];  bf0[1] = wr1[0];
    v2f bf1; bf1[0] = wr0[16]; bf1[1] = wr1[16];
    acc0 = wmma4(af, bf0, acc0);
    acc1 = wmma4(af, bf1, acc1);
  }

  // bias + residual into LDS
  {
    const int c0 = col0 + l16, c1 = col0 + 16 + l16;
    const float b0 = bo[c0], b1 = bo[c1];
    #pragma unroll
    for (int j = 0; j < 8; ++j) {
      const int row = row0 + j + 8 * g;
      X[j + 8 * g][c0] = acc0[j] + b0 + query[(size_t)row * DD + c0];
      X[j + 8 * g][c1] = acc1[j] + b1 + query[(size_t)row * DD + c1];
    }
  }
  __syncthreads();

  // LayerNorm: 16 threads per row, two-pass mean/variance (matches reference).
  const int r = tid >> 4;
  const int l = tid & 15;
  float sum = 0.0f;
  #pragma unroll
  for (int i = 0; i < 16; ++i) sum += X[r][l + 16 * i];
  #pragma unroll
  for (int off = 1; off < 16; off <<= 1) sum += __shfl_xor(sum, off, 16);
  const float mu = sum * (1.0f / 256.0f);
  float vs = 0.0f;
  #pragma unroll
  for (int i = 0; i < 16; ++i) { const float d2 = X[r][l + 16 * i] - mu; vs += d2 * d2; }
  #pragma unroll
  for (int off = 1; off < 16; off <<= 1) vs += __shfl_xor(vs, off, 16);
  const float inv = rsqrtf(vs * (1.0f / 256.0f) + 1e-3f);
  #pragma unroll
  for (int i = 0; i < 16; ++i) {
    const int c = l + 16 * i;
    out[(size_t)(row0 + r) * DD + c] = (X[r][c] - mu) * inv * gamma[c] + beta[c];
  }
}

// ---------------------------------------------------------------------------
extern "C" void kernel_launch(void* const* d_in, const int* in_sizes, int n_in,
                              void* d_out, int out_size, void* d_ws, size_t ws_size,
                              hipStream_t stream) {
  (void)in_sizes; (void)n_in; (void)out_size; (void)ws_size;
  const float* query = (const float*)d_in[0];
  const float* value = (const float*)d_in[1];
  const float* Wq    = (const float*)d_in[2];
  const float* bq    = (const float*)d_in[3];
  const float* Wk    = (const float*)d_in[4];
  const float* bk    = (const float*)d_in[5];
  const float* Wv    = (const float*)d_in[6];
  const float* bv    = (const float*)d_in[7];
  const float* Wo    = (const float*)d_in[8];
  const float* bo    = (const float*)d_in[9];
  const float* gamma = (const float*)d_in[10];
  const float* beta  = (const float*)d_in[11];

  float* ws   = (float*)d_ws;          // needs 4 * 16 MiB = 64 MiB
  float* qbuf = ws;
  float* kbuf = ws + NMAT;
  float* vbuf = ws + 2 * NMAT;
  float* ctx  = ws + 3 * NMAT;

  qkv_proj_kernel<<<dim3(NROWS / 16, 3), dim3(128), 0, stream>>>(
      query, value, Wq, bq, Wk, bk, Wv, bv, qbuf, kbuf, vbuf);
  attn_kernel<<<dim3(SS / 64, HH, BB), dim3(128), 0, stream>>>(
      qbuf, kbuf, vbuf, ctx);
  out_ln_kernel<<<dim3(NROWS / 16), dim3(256), 0, stream>>>(
      ctx, Wo, bo, query, gamma, beta, (float*)d_out);
}